// Create_Patch_33809982554293
// MI455X (gfx1250) — compile-verified
//
#include <hip/hip_runtime.h>

// ---------------------------------------------------------------------------
// ViT patch embedding for MI455X (gfx1250, wave32).
//   out[b, 0,   e] = cls[e] + pos[0, e]
//   out[b, 1+p, e] = sum_k patches[b,p,k] * W[e,k] + pos[1+p, e]
// GEMM: M = B*64 = 65536, K = 48, N = 768, exact-FP32 via
// V_WMMA_F32_16X16X4_F32 (12 chained K=4 steps per 16x16 tile).
//
// Roofline: 4.8 GFLOP vs ~217 MB HBM traffic -> write-stream bound,
// ~9.3 us floor at 23.3 TB/s. Output (204.5 MB, write-once) is stored
// non-temporal so the 192 MB L2 stays hot for x (12.6 MB, 48x reuse),
// W (147 KB) and pos_emb (200 KB).
// ---------------------------------------------------------------------------

typedef float v2f __attribute__((ext_vector_type(2)));
typedef float v4f __attribute__((ext_vector_type(4)));
typedef float v8f __attribute__((ext_vector_type(8)));

#define IMG        32
#define PATCH      4
#define PPI        64          // patches per image
#define EMBED      768
#define KDIM       48
#define BATCH      1024
#define SEQ        65
#define NTILES     (EMBED / 16)          // 48
#define MTILES     ((BATCH * PPI) / 16)  // 4096
#define WAVES_PER_N 128
#define M_ITERS    (MTILES / WAVES_PER_N)    // 32 tiles per wave (uniform)
#define TOTAL_WAVES (NTILES * WAVES_PER_N)   // 6144 -> 768 blocks of 8 waves

#define X_IMG_STRIDE   (3 * IMG * IMG)       // floats per image
#define OUT_IMG_STRIDE ((size_t)SEQ * EMBED) // floats per image of output

__global__ __launch_bounds__(256)
void vit_patch_embed_wmma(const float* __restrict__ x,     // [B,1,3,32,32]
                          const float* __restrict__ W,     // [768,48] row-major
                          const float* __restrict__ pos,   // [1,65,768]
                          float* __restrict__ out)         // [B,65,768]
{
    const int lane = threadIdx.x & 31;
    const int wave = blockIdx.x * (blockDim.x >> 5) + (threadIdx.x >> 5);
    const int half = lane >> 4;        // 0: lanes 0-15, 1: lanes 16-31
    const int l16  = lane & 15;

    const int ntile  = wave % NTILES;        // 16-wide N slice
    const int mslice = wave / NTILES;        // 0..127, M-tile phase
    const int ncol   = ntile * 16 + l16;     // lane's embed column (B/C/D role)

    // ---- B fragments: B[k][n] = W[n][k]; resident for the whole kernel. ----
    // 32-bit B 4x16 layout: lanes 0-15 hold K=0..1 (VGPR0=K0, VGPR1=K1),
    // lanes 16-31 hold K=2..3 -> per lane one contiguous float2 of W's row.
    v2f Bf[12];
    {
        const float* wrow = W + ncol * KDIM;
#pragma unroll
        for (int s = 0; s < 12; ++s)
            Bf[s] = *(const v2f*)(wrow + 4 * s + 2 * half);
    }

    // ---- pos_emb: rows depend only on mslice&3 (stride 128 % 4 == 0). ----
    const int prow_base = (mslice & 3) * 16;
    float pe[8];
#pragma unroll
    for (int j = 0; j < 8; ++j)
        pe[j] = pos[(size_t)(1 + prow_base + j + 8 * half) * EMBED + ncol];

    // ---- Loop-invariant patch geometry (mt & 3 == mslice & 3 always). ----
    const int p  = ((mslice & 3) << 4) + l16;   // patch index in image (A row = l16)
    const int gr = p >> 3, gc = p & 7;
    const int img0 = mslice >> 2;

    const float* xptr = x + (size_t)img0 * X_IMG_STRIDE
                          + (gr * PATCH) * IMG + gc * PATCH + 2 * half;
    float* optr = out + ((size_t)img0 * SEQ + 1 + prow_base) * EMBED + ncol;

    // Every wave runs exactly M_ITERS tiles: constant trip count, no EXEC
    // masking anywhere near the WMMAs.
#pragma unroll 1
    for (int it = 0; it < M_ITERS; ++it) {
        if (it + 1 < M_ITERS)   // scalar, compile-time-bounded guard
            __builtin_prefetch(xptr + (size_t)(WAVES_PER_N / 4) * X_IMG_STRIDE, 0, 3);

        v8f acc = {};
#pragma unroll
        for (int s = 0; s < 12; ++s) {
            // K = 4s + 2*half .. +1 -> channel s/4, intra-patch row s%4,
            // pw = 2*half: one contiguous float2 per lane (ISA A layout).
            const int c = s >> 2, ph = s & 3;
            v2f Af = *(const v2f*)(xptr + c * (IMG * IMG) + ph * IMG);
            acc = __builtin_amdgcn_wmma_f32_16x16x4_f32(
                /*neg_a=*/false, Af, /*neg_b=*/false, Bf[s],
                /*c_mod=*/(short)0, acc, /*reuse_a=*/false, /*reuse_b=*/false);
        }

        // D layout: lane l, VGPR j -> row m = j + 8*half, col = l%16.
        // Non-temporal: output is write-once, larger than L2.
#pragma unroll
        for (int j = 0; j < 8; ++j)
            __builtin_nontemporal_store(acc[j] + pe[j],
                                        optr + (size_t)(j + 8 * half) * EMBED);

        xptr += (size_t)(WAVES_PER_N / 4) * X_IMG_STRIDE;    // +32 images
        optr += (size_t)(WAVES_PER_N / 4) * OUT_IMG_STRIDE;  // +32 images
    }
}

// out[b, 0, :] = cls + pos[0, :], vectorized 128-bit, non-temporal stores.
__global__ __launch_bounds__(256)
void vit_cls_row(const float* __restrict__ cls,
                 const float* __restrict__ pos,
                 float* __restrict__ out)
{
    const int idx = blockIdx.x * blockDim.x + threadIdx.x;   // over B * 768/4
    const int b  = idx / (EMBED / 4);
    const int e4 = idx - b * (EMBED / 4);
    const v4f c = ((const v4f*)cls)[e4];
    const v4f q = ((const v4f*)pos)[e4];
    const v4f r = c + q;
    __builtin_nontemporal_store(r, (v4f*)(out + (size_t)b * SEQ * EMBED) + e4);
}

extern "C" void kernel_launch(void* const* d_in, const int* in_sizes, int n_in,
                              void* d_out, int out_size, void* d_ws, size_t ws_size,
                              hipStream_t stream)
{
    const float* x   = (const float*)d_in[0];   // [1024,1,3,32,32]
    const float* W   = (const float*)d_in[1];   // [768,48]
    const float* cls = (const float*)d_in[2];   // [1,768]
    const float* pos = (const float*)d_in[3];   // [1,65,768]
    float* out = (float*)d_out;                 // [1024,65,768]

    // cls rows: 1024 * 192 v4f lanes = 196608 threads.
    vit_cls_row<<<(BATCH * (EMBED / 4) + 255) / 256, 256, 0, stream>>>(cls, pos, out);

    // GEMM + pos add: 768 blocks x 8 waves = 6144 waves,
    // each wave = one 16-col N slice x 32 M tiles of 16 patches.
    vit_patch_embed_wmma<<<TOTAL_WAVES / 8, 256, 0, stream>>>(x, W, pos, out);
}